// task_3607772528871
// MI455X (gfx1250) — compile-verified
//
#include <hip/hip_runtime.h>
#include <hip/hip_bf16.h>

// ---------------------------------------------------------------------------
// Per-sample gathered-expert Linear(256 -> 1) + sigmoid, mapped onto CDNA5
// V_WMMA_F32_16X16X4_F32 (full fp32 precision; 16x16x4 matrix op).
//
// Tile = 16 samples handled by one wave32.
//   A (16x4 f32)  : rows = samples, cols = a K-chunk of x features.
//   B (4x16 f32)  : col n = the gathered expert-weight slice for sample n.
//   D = A x B + C : diagonal D[m][m] accumulates dot(x[m], W[e_m]).
//
// Feature -> K-slot mapping (identical construction for A and B so the
// contraction pairs matching features; documented layout k = vgpr + 2*half):
//   per iteration u, lane loads float4 at column (8u + 4*half); the .xy pair
//   feeds WMMA #1 (vgpr0/1), the .zw pair feeds WMMA #2.
//
// Roofline: 409.6 MB of x streamed once via b128 loads + 1.6 MB stores at
// 23.3 TB/s => ~17.7 us floor; 0.205 GFLOP of real work (0.5 FLOP/B) is
// fully hidden behind the stream. W (102 KB) is L2-resident (192 MB L2),
// so gathered B-fragment reads cost ~no HBM traffic after first touch.
// ---------------------------------------------------------------------------

typedef float v2f __attribute__((ext_vector_type(2)));
typedef float v4f __attribute__((ext_vector_type(4)));
typedef float v8f __attribute__((ext_vector_type(8)));

#define HID   256
#define TILE  16

__global__ __launch_bounds__(256) void moe_expert_sigmoid_wmma(
    const float* __restrict__ x,    // [N, 256]
    const int*   __restrict__ num,  // [N]
    const int*   __restrict__ cls,  // [100]
    const float* __restrict__ W,    // [100, 256, 1] -> rows of 256 contiguous
    const float* __restrict__ b,    // [100, 1]
    float*       __restrict__ out,  // [N, 1]
    int Ntot, int nTiles)
{
    const int lane        = threadIdx.x & 31;
    const int wave        = threadIdx.x >> 5;
    const int wavesPerBlk = blockDim.x >> 5;
    const int gwave       = blockIdx.x * wavesPerBlk + wave;
    const int nWaves      = gridDim.x * wavesPerBlk;

    const int n  = lane & 15;   // sample-in-tile (A row index / B column index)
    const int hi = lane >> 4;   // half-wave selector

    for (int tile = gwave; tile < nTiles; tile += nWaves) {
        const int s0  = tile * TILE;
        int row = s0 + n;
        if (row >= Ntot) row = Ntot - 1;           // tail clamp (EXEC stays full)

        const int e = cls[num[row]];               // expert id for this sample
        const float* __restrict__ xr = x + (size_t)row * HID;
        const float* __restrict__ wr = W + (size_t)e   * HID;

        // Prefetch this lane's row for the wave's next tile (global_prefetch_b8).
        if (tile + nWaves < nTiles) {
            __builtin_prefetch(x + (size_t)(row + nWaves * TILE) * HID, 0, 0);
        }

        v8f acc = {0.f, 0.f, 0.f, 0.f, 0.f, 0.f, 0.f, 0.f};

        #pragma unroll 4
        for (int u = 0; u < HID / 8; ++u) {
            const int col = u * 8 + hi * 4;        // 8 features per iteration
            v4f a4 = *(const v4f*)(xr + col);      // global_load_b128
            v4f b4 = *(const v4f*)(wr + col);      // global_load_b128 (L2 hit)

            v2f a0 = {a4.x, a4.y}, b0 = {b4.x, b4.y};
            v2f a1 = {a4.z, a4.w}, b1 = {b4.z, b4.w};

            // 8 args: (neg_a, A, neg_b, B, c_mod, C, reuse_a, reuse_b)
            acc = __builtin_amdgcn_wmma_f32_16x16x4_f32(
                      false, a0, false, b0, (short)0, acc, false, false);
            acc = __builtin_amdgcn_wmma_f32_16x16x4_f32(
                      false, a1, false, b1, (short)0, acc, false, false);
        }

        // --- Diagonal extraction (after all WMMAs; EXEC divergence OK now) ---
        // D[m][m]: m<8  -> lane m,     VGPR m
        //          m>=8 -> lane m+16,  VGPR m-8   (i.e. lanes 24..31)
        const bool lo  = (lane < 8);
        const bool hi8 = (lane >= 24);
        if (lo || hi8) {
            const int vsel = lo ? lane : (lane - 24);
            float r = acc[0];
            #pragma unroll
            for (int i = 1; i < 8; ++i) r = (vsel == i) ? (float)acc[i] : r;

            const int m = lo ? lane : (lane - 16);       // sample-in-tile
            const int orow = s0 + m;
            if (orow < Ntot) {
                // this lane's n == m, so its 'e' is the right expert for bias
                const float y = r + b[e];
                // fast sigmoid: v_exp_f32 + v_rcp_f32 (skip IEEE div chain)
                out[orow] = __builtin_amdgcn_rcpf(1.0f + __expf(-y));
            }
        }
    }
}

extern "C" void kernel_launch(void* const* d_in, const int* in_sizes, int n_in,
                              void* d_out, int out_size, void* d_ws, size_t ws_size,
                              hipStream_t stream) {
    const float* x   = (const float*)d_in[0];
    const int*   num = (const int*)  d_in[1];
    const int*   cls = (const int*)  d_in[2];
    const float* W   = (const float*)d_in[3];
    const float* b   = (const float*)d_in[4];
    float* out = (float*)d_out;

    const int Ntot   = in_sizes[1];               // 400000
    const int nTiles = (Ntot + TILE - 1) / TILE;  // 25000

    const int threads      = 256;                 // 8 waves per block
    const int wavesPerBlk  = threads / 32;
    int blocks = (nTiles + wavesPerBlk - 1) / wavesPerBlk;   // 3125: 1 tile/wave
    if (blocks > 16384) blocks = 16384;           // grid-stride covers the rest

    moe_expert_sigmoid_wmma<<<blocks, threads, 0, stream>>>(
        x, num, cls, W, b, out, Ntot, nTiles);
}